// BangaloreGAT_83193516524091
// MI455X (gfx1250) — compile-verified
//
#include <hip/hip_runtime.h>
#include <math.h>

typedef float v2f __attribute__((ext_vector_type(2)));
typedef float v8f __attribute__((ext_vector_type(8)));

// ---------------------------------------------------------------------------
// GEMM: C[M,N] = A[M,K] * B[K,N], fp32, one 16x16 tile per wave32 using
// V_WMMA_F32_16X16X4_F32. M,N multiples of 16; K multiple of 4.
// A 16x4 layout: lanes 0-15 -> M=l, VGPR0/1 = K=0/1 ; lanes 16-31 -> K=2/3.
// B 4x16 layout: VGPR0 = rows K=0 (lanes0-15) / K=2 (lanes16-31), VGPR1 = K=1/K=3.
// C/D layout: VGPR v -> M = v + 8*(lane>=16), N = lane&15.
// ---------------------------------------------------------------------------
__global__ void gat_gemm_wmma_f32(const float* __restrict__ A,
                                  const float* __restrict__ B,
                                  float* __restrict__ C,
                                  int M, int K, int N) {
    const int wave = (int)((blockIdx.x * blockDim.x + threadIdx.x) >> 5);
    const int lane = threadIdx.x & 31;
    const int tilesN = N >> 4;
    const int tilesM = M >> 4;
    if (wave >= tilesM * tilesN) return;           // wave-uniform: EXEC stays all-1s
    const int tm = wave / tilesN;
    const int tn = wave - tm * tilesN;
    const int half = lane >> 4;                    // 0: K pair 0,1 ; 1: K pair 2,3
    const int l    = lane & 15;

    const float* __restrict__ arow = A + (size_t)(tm * 16 + l) * (size_t)K;
    const float* __restrict__ bcol = B + (size_t)(tn * 16 + l);

    v8f acc = {};
    for (int k0 = 0; k0 < K; k0 += 4) {
        const int ka = k0 + 2 * half;
        v2f a, b;
        a.x = arow[ka];
        a.y = arow[ka + 1];
        b.x = bcol[(size_t)ka * (size_t)N];
        b.y = bcol[(size_t)(ka + 1) * (size_t)N];
        acc = __builtin_amdgcn_wmma_f32_16x16x4_f32(
            /*neg_a=*/false, a, /*neg_b=*/false, b,
            /*c_mod=*/(short)0, acc, /*reuse_a=*/false, /*reuse_b=*/false);
    }

    const int mbase = tm * 16 + 8 * half;
#pragma unroll
    for (int v = 0; v < 8; ++v)
        C[(size_t)(mbase + v) * (size_t)N + (size_t)(tn * 16 + l)] = acc[v];
}

// ---------------------------------------------------------------------------
// Per-(node,head) attention source/dest terms: s = <h, a_src>, d = <h, a_dst>
// ---------------------------------------------------------------------------
__global__ void gat_attn_sd(const float* __restrict__ h,
                            const float* __restrict__ a_src,
                            const float* __restrict__ a_dst,
                            float* __restrict__ s, float* __restrict__ d,
                            int N, int H, int C) {
    const int t = blockIdx.x * blockDim.x + threadIdx.x;
    if (t >= N * H) return;
    const int n = t / H, hh = t - n * H;
    const float* hp = h + (size_t)n * (size_t)(H * C) + (size_t)hh * C;
    const float* as = a_src + hh * C;
    const float* ad = a_dst + hh * C;
    float ss = 0.f, dd = 0.f;
    for (int c = 0; c < C; ++c) { const float v = hp[c]; ss += v * as[c]; dd += v * ad[c]; }
    s[t] = ss; d[t] = dd;
}

__device__ __forceinline__ void edge_src_dst(const int* ei, int E, int e, int& src, int& dst) {
    if (e < E) { src = ei[e]; dst = ei[E + e]; }
    else       { src = dst = e - E; }              // appended self-loops
}

__device__ __forceinline__ float leaky02(float x) { return x > 0.f ? x : 0.2f * x; }

__device__ __forceinline__ unsigned fenc(float f) {
    unsigned u = __float_as_uint(f);
    return (u & 0x80000000u) ? ~u : (u | 0x80000000u);
}
__device__ __forceinline__ float fdec(unsigned u) {
    return __uint_as_float((u & 0x80000000u) ? (u ^ 0x80000000u) : ~u);
}

// Pass A: per-destination segment max of edge logits (ordered-uint atomicMax)
__global__ void gat_edge_max(const int* __restrict__ ei, int E, int Etot,
                             const float* __restrict__ s, const float* __restrict__ d,
                             unsigned* __restrict__ menc, int H) {
    const int t = blockIdx.x * blockDim.x + threadIdx.x;
    if (t >= Etot * H) return;
    const int e = t / H, hh = t - e * H;
    int src, dst; edge_src_dst(ei, E, e, src, dst);
    const float ev = leaky02(s[src * H + hh] + d[dst * H + hh]);
    atomicMax(&menc[dst * H + hh], fenc(ev));
}

// Pass B: denom = segment_sum exp(e - m)
__global__ void gat_edge_sum(const int* __restrict__ ei, int E, int Etot,
                             const float* __restrict__ s, const float* __restrict__ d,
                             const unsigned* __restrict__ menc,
                             float* __restrict__ den, int H) {
    const int t = blockIdx.x * blockDim.x + threadIdx.x;
    if (t >= Etot * H) return;
    const int e = t / H, hh = t - e * H;
    int src, dst; edge_src_dst(ei, E, e, src, dst);
    const float ev = leaky02(s[src * H + hh] + d[dst * H + hh]);
    const float m  = fdec(menc[dst * H + hh]);
    atomicAdd(&den[dst * H + hh], expf(ev - m));
}

// Pass C: out[dst] += alpha * h[src]; one wave per edge, lanes stride channels.
__global__ void gat_edge_agg(const int* __restrict__ ei, int E, int Etot,
                             const float* __restrict__ s, const float* __restrict__ d,
                             const unsigned* __restrict__ menc,
                             const float* __restrict__ den,
                             const float* __restrict__ h,
                             float* __restrict__ out, int H, int C) {
    const int wave = (int)((blockIdx.x * blockDim.x + threadIdx.x) >> 5);
    const int lane = threadIdx.x & 31;
    if (wave >= Etot) return;
    int src, dst; edge_src_dst(ei, E, wave, src, dst);
    float w[4];
#pragma unroll
    for (int hh = 0; hh < 4; ++hh) {
        const float ev = leaky02(s[src * H + hh] + d[dst * H + hh]);
        const float m  = fdec(menc[dst * H + hh]);
        w[hh] = expf(ev - m) / (den[dst * H + hh] + 1e-16f);
    }
    const int F = H * C;
    const float* hs = h + (size_t)src * (size_t)F;
    float* od = out + (size_t)dst * (size_t)F;
    for (int c = lane; c < F; c += 32)
        atomicAdd(&od[c], w[c / C] * hs[c]);
}

// bias + BatchNorm(eval) + ELU, in place
__global__ void gat_bn_elu(float* __restrict__ x,
                           const float* __restrict__ b, const float* __restrict__ g,
                           const float* __restrict__ be, const float* __restrict__ m,
                           const float* __restrict__ v, int total, int F) {
    const int t = blockIdx.x * blockDim.x + threadIdx.x;
    if (t >= total) return;
    const int f = t % F;
    float val = x[t] + b[f];
    val = (val - m[f]) * rsqrtf(v[f] + 1e-5f) * g[f] + be[f];
    x[t] = val > 0.f ? val : (expf(val) - 1.f);
}

// final FC: out[n] = <h[n,:], w> + bias ; one wave per node, shfl reduce
__global__ void gat_final_fc(const float* __restrict__ h, const float* __restrict__ w,
                             const float* __restrict__ bptr, float* __restrict__ out,
                             int N, int F) {
    const int wave = (int)((blockIdx.x * blockDim.x + threadIdx.x) >> 5);
    const int lane = threadIdx.x & 31;
    if (wave >= N) return;
    const float* hp = h + (size_t)wave * (size_t)F;
    float acc = 0.f;
    for (int c = lane; c < F; c += 32) acc += hp[c] * w[c];
#pragma unroll
    for (int off = 16; off > 0; off >>= 1) acc += __shfl_down(acc, off, 32);
    if (lane == 0) out[wave] = acc + bptr[0];
}

// ---------------------------------------------------------------------------
extern "C" void kernel_launch(void* const* d_in, const int* in_sizes, int n_in,
                              void* d_out, int out_size, void* d_ws, size_t ws_size,
                              hipStream_t stream) {
    const float* x    = (const float*)d_in[0];
    const int*   ei   = (const int*)d_in[1];
    const float* W1   = (const float*)d_in[2];
    const float* a1s  = (const float*)d_in[3];
    const float* a1d  = (const float*)d_in[4];
    const float* b1   = (const float*)d_in[5];
    const float* g1   = (const float*)d_in[6];
    const float* be1  = (const float*)d_in[7];
    const float* m1   = (const float*)d_in[8];
    const float* v1   = (const float*)d_in[9];
    const float* W2   = (const float*)d_in[10];
    const float* a2s  = (const float*)d_in[11];
    const float* a2d  = (const float*)d_in[12];
    const float* b2   = (const float*)d_in[13];
    const float* g2   = (const float*)d_in[14];
    const float* be2  = (const float*)d_in[15];
    const float* m2   = (const float*)d_in[16];
    const float* v2   = (const float*)d_in[17];
    const float* fcW  = (const float*)d_in[18];
    const float* fcb  = (const float*)d_in[19];
    float* out = (float*)d_out;

    const int N    = out_size;              // 50000
    const int E    = in_sizes[1] / 2;       // 800000
    const int Etot = E + N;                 // + self loops
    const int Fin  = in_sizes[0] / N;       // 128
    const int H    = 4;
    const int F1   = in_sizes[5];           // 256
    const int F2   = in_sizes[13];          // 128
    const int C1   = F1 / H;                // 64
    const int C2   = F2 / H;                // 32

    // workspace layout (agg2 reuses h1 region; F2 <= F1)
    char* ws = (char*)d_ws;
    float*    h1   = (float*)ws;                         ws += (size_t)N * F1 * sizeof(float);
    float*    hact = (float*)ws;                         ws += (size_t)N * F1 * sizeof(float);
    float*    h2   = (float*)ws;                         ws += (size_t)N * F2 * sizeof(float);
    float*    sbuf = (float*)ws;                         ws += (size_t)N * H * sizeof(float);
    float*    dbuf = (float*)ws;                         ws += (size_t)N * H * sizeof(float);
    unsigned* menc = (unsigned*)ws;                      ws += (size_t)N * H * sizeof(unsigned);
    float*    den  = (float*)ws;                         ws += (size_t)N * H * sizeof(float);
    float*    agg2 = h1;                                 // reuse after layer 1

    const int BLK = 256;
    const int edgeHThreads = Etot * H;
    const int edgeHBlocks  = (edgeHThreads + BLK - 1) / BLK;
    const int aggBlocks    = ((Etot * 32) + BLK - 1) / BLK;
    const int nhBlocks     = ((N * H) + BLK - 1) / BLK;

    // ---------------- Layer 1 ----------------
    {
        const int waves = (N / 16) * (F1 / 16);
        gat_gemm_wmma_f32<<<(waves * 32 + BLK - 1) / BLK, BLK, 0, stream>>>(x, W1, h1, N, Fin, F1);
    }
    gat_attn_sd<<<nhBlocks, BLK, 0, stream>>>(h1, a1s, a1d, sbuf, dbuf, N, H, C1);

    hipMemsetAsync(menc, 0, (size_t)N * H * sizeof(unsigned), stream);
    gat_edge_max<<<edgeHBlocks, BLK, 0, stream>>>(ei, E, Etot, sbuf, dbuf, menc, H);

    hipMemsetAsync(den, 0, (size_t)N * H * sizeof(float), stream);
    gat_edge_sum<<<edgeHBlocks, BLK, 0, stream>>>(ei, E, Etot, sbuf, dbuf, menc, den, H);

    hipMemsetAsync(hact, 0, (size_t)N * F1 * sizeof(float), stream);
    gat_edge_agg<<<aggBlocks, BLK, 0, stream>>>(ei, E, Etot, sbuf, dbuf, menc, den, h1, hact, H, C1);

    gat_bn_elu<<<((N * F1) + BLK - 1) / BLK, BLK, 0, stream>>>(hact, b1, g1, be1, m1, v1, N * F1, F1);

    // ---------------- Layer 2 ----------------
    {
        const int waves = (N / 16) * (F2 / 16);
        gat_gemm_wmma_f32<<<(waves * 32 + BLK - 1) / BLK, BLK, 0, stream>>>(hact, W2, h2, N, F1, F2);
    }
    gat_attn_sd<<<nhBlocks, BLK, 0, stream>>>(h2, a2s, a2d, sbuf, dbuf, N, H, C2);

    hipMemsetAsync(menc, 0, (size_t)N * H * sizeof(unsigned), stream);
    gat_edge_max<<<edgeHBlocks, BLK, 0, stream>>>(ei, E, Etot, sbuf, dbuf, menc, H);

    hipMemsetAsync(den, 0, (size_t)N * H * sizeof(float), stream);
    gat_edge_sum<<<edgeHBlocks, BLK, 0, stream>>>(ei, E, Etot, sbuf, dbuf, menc, den, H);

    hipMemsetAsync(agg2, 0, (size_t)N * F2 * sizeof(float), stream);
    gat_edge_agg<<<aggBlocks, BLK, 0, stream>>>(ei, E, Etot, sbuf, dbuf, menc, den, h2, agg2, H, C2);

    gat_bn_elu<<<((N * F2) + BLK - 1) / BLK, BLK, 0, stream>>>(agg2, b2, g2, be2, m2, v2, N * F2, F2);

    // ---------------- Readout ----------------
    gat_final_fc<<<(N * 32 + BLK - 1) / BLK, BLK, 0, stream>>>(agg2, fcW, fcb, out, N, F2);
}